// NoiseAwareQuantumLayer_30837865185839
// MI455X (gfx1250) — compile-verified
//
#include <hip/hip_runtime.h>
#include <hip/hip_bf16.h>

// ---------------------------------------------------------------------------
// 12-qubit batched state-vector sim, LDS-resident per sample, plus a WMMA
// (V_WMMA_F32_16X16X4_F32) tail for the readout-mitigation affine map.
//
// Math reduction used:  M = ones(12,12) - eye(12)  =>  pinv(M) = J/11 - I
//   out[b,i] = 2*(((exps+1)/2) @ (J/11 - I))[b,i] - 1
//            = (exps @ (J/11 - I))[b,i] - 10/11
// ---------------------------------------------------------------------------

#define NQ   12
#define DIM  4096           // 2^12
#define NTHR 256

typedef __attribute__((ext_vector_type(2))) float v2f;
typedef __attribute__((ext_vector_type(8))) float v8f;

// ---------------------------------------------------------------------------
// Kernel 1: per-sample state-vector simulation. One block = one sample.
// State (4096 complex64 = 32KB) lives in LDS. 256 threads = 8 wave32s.
// Writes exps[b*12 + i] = <Z_i> to workspace.
// ---------------------------------------------------------------------------
__global__ __launch_bounds__(NTHR) void qsim_kernel(const float* __restrict__ x,
                                                    const float* __restrict__ params,
                                                    int n_layers,
                                                    float* __restrict__ exps_out)
{
    __shared__ float2 st[DIM];           // 32 KB state
    __shared__ float  red[12 * NTHR];    // 12 KB reduction buffer

    const int b   = blockIdx.x;
    const int tid = threadIdx.x;

    // |0...0> init
    for (int k = tid; k < DIM; k += NTHR)
        st[k] = make_float2(k == 0 ? 1.0f : 0.0f, 0.0f);
    __syncthreads();

    // ---- Angle embedding: RX(x[b,q]) on qubit q (bit NQ-1-q of index) ----
    for (int q = 0; q < NQ; ++q) {
        const float xv = x[b * NQ + q];
        float sh, ch;
        sincosf(0.5f * xv, &sh, &ch);
        const int bit  = NQ - 1 - q;
        const int strd = 1 << bit;
        for (int p = tid; p < DIM / 2; p += NTHR) {
            const int i0 = ((p >> bit) << (bit + 1)) | (p & (strd - 1));
            const int i1 = i0 | strd;
            const float2 a0 = st[i0];
            const float2 a1 = st[i1];
            // g = [[c, -i s], [-i s, c]]
            st[i0] = make_float2(ch * a0.x + sh * a1.y, ch * a0.y - sh * a1.x);
            st[i1] = make_float2(sh * a0.y + ch * a1.x, ch * a1.y - sh * a0.x);
        }
        __syncthreads();
    }

    // ---- Variational layers: Rot gates + CNOT chain ----
    for (int l = 0; l < n_layers; ++l) {
        for (int q = 0; q < NQ; ++q) {
            const float* pp  = params + (l * NQ + q) * 3;
            const float phi  = pp[0], th = pp[1], om = pp[2];
            float stt, ct, sa, ca, sb, cb;
            sincosf(0.5f * th, &stt, &ct);
            sincosf(0.5f * (phi + om), &sa, &ca);
            sincosf(0.5f * (phi - om), &sb, &cb);
            // g00 = e^{-i(phi+om)/2} ct ; g01 = -e^{i(phi-om)/2} st
            // g10 = e^{-i(phi-om)/2} st ; g11 = e^{ i(phi+om)/2} ct
            const float g00r =  ca * ct,  g00i = -sa * ct;
            const float g01r = -cb * stt, g01i = -sb * stt;
            const float g10r =  cb * stt, g10i = -sb * stt;
            const float g11r =  ca * ct,  g11i =  sa * ct;

            const int bit  = NQ - 1 - q;
            const int strd = 1 << bit;
            for (int p = tid; p < DIM / 2; p += NTHR) {
                const int i0 = ((p >> bit) << (bit + 1)) | (p & (strd - 1));
                const int i1 = i0 | strd;
                const float2 a0 = st[i0];
                const float2 a1 = st[i1];
                st[i0] = make_float2(
                    g00r * a0.x - g00i * a0.y + g01r * a1.x - g01i * a1.y,
                    g00r * a0.y + g00i * a0.x + g01r * a1.y + g01i * a1.x);
                st[i1] = make_float2(
                    g10r * a0.x - g10i * a0.y + g11r * a1.x - g11i * a1.y,
                    g10r * a0.y + g10i * a0.x + g11r * a1.y + g11i * a1.x);
            }
            __syncthreads();
        }
        // CNOT chain: control q (bit tb+1), target q+1 (bit tb). Swap
        // amplitudes (c=1,t=0) <-> (c=1,t=1).
        for (int q = 0; q < NQ - 1; ++q) {
            const int tb = NQ - 2 - q;           // target bit; control = tb+1
            for (int p = tid; p < DIM / 4; p += NTHR) {
                const int low  = p & ((1 << tb) - 1);
                const int high = p >> tb;
                const int i1 = (high << (tb + 2)) | (1 << (tb + 1)) | low;
                const int i2 = i1 | (1 << tb);
                const float2 t = st[i1];
                st[i1] = st[i2];
                st[i2] = t;
            }
            __syncthreads();
        }
    }

    // ---- Expectations <Z_i> = sum_k |amp_k|^2 * (1 - 2*bit_{11-i}(k)) ----
    float acc[12];
#pragma unroll
    for (int i = 0; i < 12; ++i) acc[i] = 0.0f;
    for (int k = tid; k < DIM; k += NTHR) {
        const float2 a = st[k];
        const float pr = a.x * a.x + a.y * a.y;
#pragma unroll
        for (int i = 0; i < 12; ++i)
            acc[i] += ((k >> (11 - i)) & 1) ? -pr : pr;
    }
#pragma unroll
    for (int i = 0; i < 12; ++i) red[i * NTHR + tid] = acc[i];
    __syncthreads();
    for (int off = NTHR / 2; off > 0; off >>= 1) {
        if (tid < off) {
#pragma unroll
            for (int i = 0; i < 12; ++i)
                red[i * NTHR + tid] += red[i * NTHR + tid + off];
        }
        __syncthreads();
    }
    if (tid < 12)
        exps_out[b * 12 + tid] = red[tid * NTHR];
}

// ---------------------------------------------------------------------------
// Kernel 2: readout mitigation as an affine map via V_WMMA_F32_16X16X4_F32.
//   out = exps @ T - 10/11,  T[j][i] = (i==j ? 1/11 - 1 : 1/11), cols 12..15 = 1/11 (unused)
// One wave32 per block, 16 samples per block. K=12 done as 3 accumulating
// K=4 WMMAs. EXEC is all ones (no divergence before the WMMAs).
//
// f32 16x16x4 VGPR striping (ISA 7.12.2):
//   A (16x4): lane m=L%16 -> a.x = A[m][2*(L/16)+0], a.y = A[m][2*(L/16)+1]
//   B (4x16): b.x = B[2*(L/16)+0][n], b.y = B[2*(L/16)+1][n], n = L%16
//   D (16x16): VGPR v, lanes 0-15 -> row v, lanes 16-31 -> row v+8, col = L%16
// ---------------------------------------------------------------------------
__global__ __launch_bounds__(32) void mitigate_wmma_kernel(const float* __restrict__ exps,
                                                           float* __restrict__ out)
{
    const int lane = threadIdx.x;        // 0..31
    const int base = blockIdx.x * 16;    // first sample of this tile
    const int m    = lane & 15;          // A row / D column index
    const int half = lane >> 4;          // 0 or 1

    const float inv11 = 1.0f / 11.0f;

    v8f c = {0.f, 0.f, 0.f, 0.f, 0.f, 0.f, 0.f, 0.f};

#pragma unroll
    for (int kk = 0; kk < 12; kk += 4) {
        const int ka = kk + 2 * half;    // K index of a.x / b.x
        v2f a, bv;
        a.x = exps[(base + m) * 12 + ka];
        a.y = exps[(base + m) * 12 + ka + 1];
        // T rows ka, ka+1 at column n = m (columns 12..15 unused: never stored)
        bv.x = (m == ka)     ? (inv11 - 1.0f) : inv11;
        bv.y = (m == ka + 1) ? (inv11 - 1.0f) : inv11;
        c = __builtin_amdgcn_wmma_f32_16x16x4_f32(
                /*neg_a=*/false, a, /*neg_b=*/false, bv,
                /*c_mod=*/(short)0, c, /*reuse_a=*/false, /*reuse_b=*/false);
    }

#pragma unroll
    for (int v = 0; v < 8; ++v) {
        const int row = v + 8 * half;    // sample within tile
        if (m < 12)
            out[(base + row) * 12 + m] = c[v] - (10.0f / 11.0f);
    }
}

// ---------------------------------------------------------------------------
extern "C" void kernel_launch(void* const* d_in, const int* in_sizes, int n_in,
                              void* d_out, int out_size, void* d_ws, size_t ws_size,
                              hipStream_t stream)
{
    const float* x      = (const float*)d_in[0];   // (B, 12) f32
    const float* params = (const float*)d_in[1];   // (L, 12, 3) f32
    float*       out    = (float*)d_out;           // (B, 12) f32
    float*       exps   = (float*)d_ws;            // (B, 12) f32 scratch

    const int B        = in_sizes[0] / NQ;         // 768
    const int n_layers = in_sizes[1] / (NQ * 3);   // 6

    qsim_kernel<<<B, NTHR, 0, stream>>>(x, params, n_layers, exps);
    mitigate_wmma_kernel<<<B / 16, 32, 0, stream>>>(exps, out);
}